// APIHyperInputLayer_8186207666461
// MI455X (gfx1250) — compile-verified
//
#include <hip/hip_runtime.h>

#define N_AGENTS  10
#define N_ENEMIES 11
#define ALLY_F    48
#define ENEMY_F   32
#define HYPER     64
#define OUT_DIM   128
#define BATCH     2048

typedef __attribute__((ext_vector_type(2))) float v2f;
typedef __attribute__((ext_vector_type(8))) float v8f;
typedef int v4i_vs __attribute__((vector_size(16)));  // matches builtin param

__device__ __forceinline__ v8f wmma_f32_16x16x4(v2f a, v2f b, v8f c) {
  return __builtin_amdgcn_wmma_f32_16x16x4_f32(false, a, false, b, (short)0, c,
                                               false, false);
}

// --- CDNA5 async global->LDS copy (ASYNCcnt-tracked), with sync fallback ---
__device__ __forceinline__ void async_copy_b128(float* lds_dst,
                                                const float* gsrc) {
#if __has_builtin(__builtin_amdgcn_global_load_async_to_lds_b128)
  __builtin_amdgcn_global_load_async_to_lds_b128(
      (__attribute__((address_space(1))) v4i_vs*)gsrc,
      (__attribute__((address_space(3))) v4i_vs*)lds_dst, 0, 0);
#else
  *(float4*)lds_dst = *(const float4*)gsrc;
#endif
}

__device__ __forceinline__ void wait_async0() {
#if __has_builtin(__builtin_amdgcn_s_wait_asynccnt)
  __builtin_amdgcn_s_wait_asynccnt(0);
#else
  asm volatile("s_wait_asynccnt 0x0" ::: "memory");
#endif
}

// Stage F*128 floats (one W2_k slice or b2) into an LDS buffer: each of the
// 256 threads issues (F*128/4)/256 async b128 transfers.
template <int F>
__device__ __forceinline__ void stage_slice_async(float* lds_dst,
                                                  const float* gsrc, int tid) {
#pragma unroll
  for (int i = 0; i < (F * OUT_DIM / 4) / 256; ++i) {
    int idx = (tid + i * 256) * 4;
    async_copy_b128(lds_dst + idx, gsrc + idx);
  }
}

// ---------------------------------------------------------------------------
// Kernel H: h = prelu(x @ W1 + b1).  Tiny fraction of total FLOPs -> VALU.
// ---------------------------------------------------------------------------
template <int F>
__global__ void hyper_kernel(const float* __restrict__ feats,
                             const float* __restrict__ W1,
                             const float* __restrict__ b1,
                             const float* __restrict__ alpha,
                             float* __restrict__ h_out, int rows) {
  int t   = blockIdx.x * blockDim.x + threadIdx.x;
  int row = t >> 6;
  int k   = t & (HYPER - 1);
  if (row >= rows) return;
  const float* fr = feats + (size_t)row * F;
  float s = b1[k];
#pragma unroll
  for (int f = 0; f < F; ++f) s = fmaf(fr[f], W1[f * HYPER + k], s);
  float a = alpha[k];
  h_out[(size_t)row * HYPER + k] = (s >= 0.f) ? s : a * s;
}

// ---------------------------------------------------------------------------
// Kernel M: per-batch hyper-GEMM, WMMA f32 16x16x4, double-buffered async
// W2_k staging (ASYNCcnt pipeline), two interleaved accumulator chains.
// ---------------------------------------------------------------------------
template <int F, int NPER>
__global__ __launch_bounds__(256) void branch_main_kernel(
    const float* __restrict__ feats, const float* __restrict__ hbuf,
    const float* __restrict__ W2, const float* __restrict__ b2,
    float* __restrict__ partial) {
  __shared__ float w2s[2][F * OUT_DIM];   // double buffer: 48KB / 32KB
  __shared__ float hs[HYPER * 16];        // h tile, transposed [k][m]

  const int tid  = threadIdx.x;
  const int lane = tid & 31;
  const int wave = tid >> 5;
  const int m    = lane & 15;             // A: M row; B: N col (within tile)
  const int kk   = (lane >> 4) * 2;       // K sub-offset for this half-wave
  const int b    = blockIdx.x;

  // Prologue: kick off async stages of b2 (buf0) and W2_0 (buf1).
  stage_slice_async<F>(&w2s[0][0], b2, tid);
  stage_slice_async<F>(&w2s[1][0], W2, tid);

  // --- per-lane A fragments: feats row m, zero-padded past NPER ------------
  v2f fr[F / 4];
  {
    const float* frow = feats + (size_t)(b * NPER + m) * F;
#pragma unroll
    for (int s = 0; s < F / 4; ++s) {
      float x = 0.f, y = 0.f;
      if (m < NPER) {
        x = frow[4 * s + kk];
        y = frow[4 * s + kk + 1];
      }
      fr[s].x = x;
      fr[s].y = y;
    }
  }

  // --- stage h tile (transposed -> conflict-free broadcast reads) ----------
#pragma unroll
  for (int i = 0; i < (HYPER * 16) / 256; ++i) {
    int idx = tid + i * 256;
    int k   = idx >> 4;
    int r   = idx & 15;
    hs[idx] = (r < NPER) ? hbuf[(size_t)(b * NPER + r) * HYPER + k] : 0.f;
  }

  wait_async0();
  __syncthreads();

  // --- bias pass: acc = X @ B2 (from buf0) ---------------------------------
  v8f acc0 = {}, acc1 = {};
  const int col = wave * 16 + m;
#pragma unroll
  for (int s = 0; s < F / 4; ++s) {
    v2f bb;
    bb.x = w2s[0][(4 * s + kk) * OUT_DIM + col];
    bb.y = w2s[0][(4 * s + kk + 1) * OUT_DIM + col];
    if (s & 1)
      acc1 = wmma_f32_16x16x4(fr[s], bb, acc1);
    else
      acc0 = wmma_f32_16x16x4(fr[s], bb, acc0);
  }

  // --- main loop: acc += (h[:,k]*X) @ W2_k, software-pipelined -------------
  for (int k = 0; k < HYPER; ++k) {
    const int cur = (k + 1) & 1;  // buffer holding W2_k
    const int nxt = k & 1;        // buffer just drained -> prefetch target
    wait_async0();                // W2_k landed (own chunks)
    __syncthreads();              // all waves' chunks visible; nxt fully read
    if (k + 1 < HYPER)
      stage_slice_async<F>(&w2s[nxt][0], W2 + (size_t)(k + 1) * F * OUT_DIM,
                           tid);
    const float sc = hs[k * 16 + m];  // h[m,k]: same M for all of lane's A
#pragma unroll
    for (int s = 0; s < F / 4; ++s) {
      v2f aa;
      aa.x = fr[s].x * sc;
      aa.y = fr[s].y * sc;
      v2f bb;
      bb.x = w2s[cur][(4 * s + kk) * OUT_DIM + col];
      bb.y = w2s[cur][(4 * s + kk + 1) * OUT_DIM + col];
      if (s & 1)
        acc1 = wmma_f32_16x16x4(aa, bb, acc1);
      else
        acc0 = wmma_f32_16x16x4(aa, bb, acc0);
    }
  }

  // --- epilogue: reduce accumulator over M (= sum over batch entities) -----
  float s8 = 0.f;
#pragma unroll
  for (int j = 0; j < 8; ++j) s8 += acc0[j] + acc1[j];
  s8 += __shfl_xor(s8, 16, 32);  // combine the two M halves (same column)
  if (lane < 16) partial[(size_t)b * OUT_DIM + col] = s8;
}

// ---------------------------------------------------------------------------
__global__ void add_kernel(const float* __restrict__ a,
                           const float* __restrict__ b,
                           float* __restrict__ out, int n) {
  int i = blockIdx.x * blockDim.x + threadIdx.x;
  if (i < n) out[i] = a[i] + b[i];
}

// ---------------------------------------------------------------------------
extern "C" void kernel_launch(void* const* d_in, const int* in_sizes, int n_in,
                              void* d_out, int out_size, void* d_ws,
                              size_t ws_size, hipStream_t stream) {
  const float* ally  = (const float*)d_in[0];
  const float* enemy = (const float*)d_in[1];
  const float* Wa1   = (const float*)d_in[2];
  const float* ba1   = (const float*)d_in[3];
  const float* aa    = (const float*)d_in[4];
  const float* Wa2   = (const float*)d_in[5];
  const float* ba2   = (const float*)d_in[6];
  const float* We1   = (const float*)d_in[7];
  const float* be1   = (const float*)d_in[8];
  const float* ae    = (const float*)d_in[9];
  const float* We2   = (const float*)d_in[10];
  const float* be2   = (const float*)d_in[11];
  float*       out   = (float*)d_out;

  const int rows_a = BATCH * N_AGENTS;   // 20480
  const int rows_e = BATCH * N_ENEMIES;  // 22528

  float* h_a = (float*)d_ws;
  float* h_e = h_a + (size_t)rows_a * HYPER;
  float* pa  = h_e + (size_t)rows_e * HYPER;
  float* pe  = pa + (size_t)BATCH * OUT_DIM;

  hyper_kernel<ALLY_F><<<(rows_a * HYPER + 255) / 256, 256, 0, stream>>>(
      ally, Wa1, ba1, aa, h_a, rows_a);
  hyper_kernel<ENEMY_F><<<(rows_e * HYPER + 255) / 256, 256, 0, stream>>>(
      enemy, We1, be1, ae, h_e, rows_e);

  branch_main_kernel<ALLY_F, N_AGENTS><<<BATCH, 256, 0, stream>>>(
      ally, h_a, Wa2, ba2, pa);
  branch_main_kernel<ENEMY_F, N_ENEMIES><<<BATCH, 256, 0, stream>>>(
      enemy, h_e, We2, be2, pe);

  const int n = BATCH * OUT_DIM;
  add_kernel<<<(n + 255) / 256, 256, 0, stream>>>(pa, pe, out, n);
}